// Diffusion_22222160789669
// MI455X (gfx1250) — compile-verified
//
#include <hip/hip_runtime.h>

// Problem constants (from reference).
constexpr int NN     = 50000;   // nodes
constexpr int DF     = 256;     // feature columns
constexpr int DF4    = DF / 4;  // float4 per row
constexpr int NE     = 400000;  // edges
constexpr int NSEG   = 4;
constexpr int NTERMS = 8;
constexpr float DIFF_T = 1.0f;

constexpr int ROWS_PER_BLK = 8;    // one wave32 per row
constexpr int MV_THREADS   = 256;  // 8 waves
constexpr int ECHUNK       = 256;  // staged edges per chunk

// Generic pointer -> raw LDS byte offset (flat LDS addr low 32 bits are the
// in-allocation offset on gfx1250).
typedef __attribute__((address_space(3))) const char lds_cchar;
__device__ __forceinline__ unsigned lds_off_of(const void* p) {
  return (unsigned)(unsigned long long)(lds_cchar*)p;
}

__global__ void zero_u32_kernel(unsigned* p, int n) {
  int i = blockIdx.x * blockDim.x + threadIdx.x;
  if (i < n) p[i] = 0u;
}

__global__ void hist_kernel(const int* __restrict__ rows,
                            unsigned* __restrict__ counts, int e_cnt) {
  int e = blockIdx.x * blockDim.x + threadIdx.x;
  if (e < e_cnt) atomicAdd(&counts[rows[e]], 1u);
}

// Single-block chunked exclusive scan over NN counts -> row_start[NN+1].
__global__ void scan_kernel(const unsigned* __restrict__ counts,
                            unsigned* __restrict__ row_start, int n) {
  __shared__ unsigned s[256];
  __shared__ unsigned carry_s;
  int tid = threadIdx.x;
  if (tid == 0) carry_s = 0u;
  __syncthreads();
  for (int base = 0; base < n; base += 256) {
    int i = base + tid;
    unsigned v = (i < n) ? counts[i] : 0u;
    s[tid] = v;
    __syncthreads();
    for (int off = 1; off < 256; off <<= 1) {
      unsigned t = (tid >= off) ? s[tid - off] : 0u;
      __syncthreads();
      s[tid] += t;
      __syncthreads();
    }
    unsigned carry = carry_s;
    if (i < n) row_start[i] = carry + s[tid] - v;  // exclusive
    __syncthreads();
    if (tid == 255) carry_s = carry + s[255];
    __syncthreads();
  }
  if (tid == 0) row_start[n] = carry_s;
}

__global__ void fill_kernel(const int* __restrict__ rows,
                            const int* __restrict__ cols,
                            const float* __restrict__ vals,
                            const unsigned* __restrict__ row_start,
                            unsigned* __restrict__ cursor,
                            int* __restrict__ csr_col,
                            float* __restrict__ csr_val, int e_cnt) {
  int e = blockIdx.x * blockDim.x + threadIdx.x;
  if (e < e_cnt) {
    int r = rows[e];
    unsigned p = row_start[r] + atomicAdd(&cursor[r], 1u);
    csr_col[p] = cols[e];
    csr_val[p] = vals[e];
  }
}

__global__ void deg_kernel(const unsigned* __restrict__ row_start,
                           const float* __restrict__ csr_val,
                           float* __restrict__ deg, int n) {
  int r = blockIdx.x * blockDim.x + threadIdx.x;
  if (r < n) {
    float s = 0.f;
    unsigned e1 = row_start[r + 1];
    for (unsigned u = row_start[r]; u < e1; ++u) s += csr_val[u];
    deg[r] = s;
  }
}

__global__ void init_kernel(const float4* __restrict__ x,
                            float4* __restrict__ A,
                            float4* __restrict__ acc, int n4) {
  int i = blockIdx.x * blockDim.x + threadIdx.x;
  if (i < n4) { float4 v = x[i]; A[i] = v; acc[i] = v; }
}

__global__ void copy_kernel(const float4* __restrict__ src,
                            float4* __restrict__ dst, int n4) {
  int i = blockIdx.x * blockDim.x + threadIdx.x;
  if (i < n4) dst[i] = src[i];
}

// 8 rows per block, one wave32 per row, each lane owns 8 feature columns
// (two float4s, lane-strided -> 512B contiguous gathers per wave).
// Rows of a block are contiguous in CSR, so one async-to-LDS stage feeds all
// 8 waves.  Fuses: B = coef*(deg[r]*A[r] - W A[r]), ACC += B.
__global__ __launch_bounds__(MV_THREADS) void matvec_kernel(
    const float* __restrict__ A, float* __restrict__ B, float* __restrict__ ACC,
    const int* __restrict__ csr_col, const float* __restrict__ csr_val,
    const unsigned* __restrict__ row_start, const float* __restrict__ deg,
    float coef) {
  __shared__ unsigned ebuf[2 * ECHUNK];  // [0,ECHUNK): cols, [ECHUNK,2*ECHUNK): vals
  const int tid  = threadIdx.x;
  const int wid  = tid >> 5;   // wave id = row within block
  const int lane = tid & 31;
  const int r0   = blockIdx.x * ROWS_PER_BLK;
  const int r    = r0 + wid;

  const int cb0 = (int)row_start[r0];                  // block's combined CSR range
  const int cb1 = (int)row_start[r0 + ROWS_PER_BLK];
  const int s0  = (int)row_start[r];                   // this wave's row range
  const int s1  = (int)row_start[r + 1];

  float4 acc0 = {0.f, 0.f, 0.f, 0.f};
  float4 acc1 = {0.f, 0.f, 0.f, 0.f};
  const float4* A4 = (const float4*)A;

  const int total = cb1 - cb0;   // uniform per block -> barriers below are safe
  if (total > 0) {
    const unsigned off_c = lds_off_of(&ebuf[tid]);
    const unsigned off_v = lds_off_of(&ebuf[ECHUNK + tid]);
    for (int base = 0; base < total; base += ECHUNK) {
      const int cbb = cb0 + base;
      int m = total - base; if (m > ECHUNK) m = ECHUNK;
      // stage m edges (cols+vals) into LDS via CDNA5 async loads
      {
        int l = (tid < m) ? tid : (m - 1);             // clamp OOB lanes in-range
        const void* gc = (const void*)(csr_col + (cbb + l));
        const void* gv = (const void*)(csr_val + (cbb + l));
        asm volatile("global_load_async_to_lds_b32 %0, %1, off"
                     :: "v"(off_c), "v"(gc) : "memory");
        asm volatile("global_load_async_to_lds_b32 %0, %1, off"
                     :: "v"(off_v), "v"(gv) : "memory");
        asm volatile("s_wait_asynccnt 0" ::: "memory");
      }
      __syncthreads();
      // this wave consumes the intersection of its row range with this chunk
      int lo = s0 > cbb ? s0 : cbb;
      int hi = s1 < cbb + m ? s1 : cbb + m;
      for (int e = lo; e < hi; ++e) {
        const int j = e - cbb;
        const int col = (int)ebuf[j];                  // LDS broadcast reads
        const float w = __uint_as_float(ebuf[ECHUNK + j]);
        const size_t rb = (size_t)col * DF4;
        float4 v0 = A4[rb + lane];                     // 512B contiguous / wave
        float4 v1 = A4[rb + 32 + lane];
        acc0.x = fmaf(w, v0.x, acc0.x); acc0.y = fmaf(w, v0.y, acc0.y);
        acc0.z = fmaf(w, v0.z, acc0.z); acc0.w = fmaf(w, v0.w, acc0.w);
        acc1.x = fmaf(w, v1.x, acc1.x); acc1.y = fmaf(w, v1.y, acc1.y);
        acc1.z = fmaf(w, v1.z, acc1.z); acc1.w = fmaf(w, v1.w, acc1.w);
      }
      __syncthreads();
    }
  }

  const float dg = deg[r];
  const size_t idx0 = (size_t)r * DF4 + lane;
  const size_t idx1 = idx0 + 32;
  float4 a0 = A4[idx0], a1 = A4[idx1];
  float4 b0, b1;
  b0.x = coef * fmaf(dg, a0.x, -acc0.x);
  b0.y = coef * fmaf(dg, a0.y, -acc0.y);
  b0.z = coef * fmaf(dg, a0.z, -acc0.z);
  b0.w = coef * fmaf(dg, a0.w, -acc0.w);
  b1.x = coef * fmaf(dg, a1.x, -acc1.x);
  b1.y = coef * fmaf(dg, a1.y, -acc1.y);
  b1.z = coef * fmaf(dg, a1.z, -acc1.z);
  b1.w = coef * fmaf(dg, a1.w, -acc1.w);
  ((float4*)B)[idx0] = b0;
  ((float4*)B)[idx1] = b1;
  float4 c0 = ((float4*)ACC)[idx0];
  float4 c1 = ((float4*)ACC)[idx1];
  c0.x += b0.x; c0.y += b0.y; c0.z += b0.z; c0.w += b0.w;
  c1.x += b1.x; c1.y += b1.y; c1.z += b1.z; c1.w += b1.w;
  ((float4*)ACC)[idx0] = c0;
  ((float4*)ACC)[idx1] = c1;
}

extern "C" void kernel_launch(void* const* d_in, const int* in_sizes, int n_in,
                              void* d_out, int out_size, void* d_ws, size_t ws_size,
                              hipStream_t stream) {
  const float* x   = (const float*)d_in[0];
  const int* erow  = (const int*)d_in[1];
  const int* ecol  = (const int*)d_in[2];
  const float* evl = (const float*)d_in[3];
  float* ACC = (float*)d_out;  // Taylor accumulator lives in the output buffer

  // Workspace layout (~106.5 MB): A | B | csr_col | csr_val | counts | row_start | deg
  float* A = (float*)d_ws;
  float* B = A + (size_t)NN * DF;
  int* csr_col   = (int*)(B + (size_t)NN * DF);
  float* csr_val = (float*)(csr_col + NE);
  unsigned* counts    = (unsigned*)(csr_val + NE);
  unsigned* row_start = counts + NN;
  float* deg = (float*)(row_start + NN + 1);

  const int TB = 256;
  // --- Build CSR + degrees on device (once per launch) ---
  zero_u32_kernel<<<(NN + TB - 1) / TB, TB, 0, stream>>>(counts, NN);
  hist_kernel<<<(NE + TB - 1) / TB, TB, 0, stream>>>(erow, counts, NE);
  scan_kernel<<<1, 256, 0, stream>>>(counts, row_start, NN);
  zero_u32_kernel<<<(NN + TB - 1) / TB, TB, 0, stream>>>(counts, NN);  // reuse as cursor
  fill_kernel<<<(NE + TB - 1) / TB, TB, 0, stream>>>(erow, ecol, evl, row_start,
                                                     counts, csr_col, csr_val, NE);
  deg_kernel<<<(NN + TB - 1) / TB, TB, 0, stream>>>(row_start, csr_val, deg, NN);

  // --- Scaling + truncated Taylor: out = [sum_k (-t/4 L)^k/k!]^4 x ---
  const int n4 = NN * DF4;
  init_kernel<<<(n4 + TB - 1) / TB, TB, 0, stream>>>((const float4*)x,
                                                     (float4*)A, (float4*)ACC, n4);
  const float t_seg = DIFF_T / (float)NSEG;
  float* cur = A;
  float* nxt = B;
  const int mv_blocks = NN / ROWS_PER_BLK;  // 6250
  for (int seg = 0; seg < NSEG; ++seg) {
    if (seg > 0)  // term_0 of next segment = current accumulator
      copy_kernel<<<(n4 + TB - 1) / TB, TB, 0, stream>>>((const float4*)ACC,
                                                         (float4*)cur, n4);
    for (int k = 1; k <= NTERMS; ++k) {
      float coef = -t_seg / (float)k;
      matvec_kernel<<<mv_blocks, MV_THREADS, 0, stream>>>(
          cur, nxt, ACC, csr_col, csr_val, row_start, deg, coef);
      float* t = cur; cur = nxt; nxt = t;
    }
  }
  (void)in_sizes; (void)n_in; (void)out_size; (void)ws_size;
}